// GMSDRCell_79456894976616
// MI455X (gfx1250) — compile-verified
//
#include <hip/hip_runtime.h>
#include <hip/hip_bf16.h>

typedef __attribute__((ext_vector_type(2))) float v2f;
typedef __attribute__((ext_vector_type(8))) float v8f;

#define Nn 4096      // nodes
#define Dd 64        // units
#define Bb 32        // batch
#define Ee 131072    // edges
#define BN 131072    // B*N rows
#define FP 132       // padded feature dim (130 -> 132, K mult of 4)
#define MO 192       // fused GEMM1 output cols: [A0 | A1 | A2']

// ---------------- small prep kernels ----------------

// Wcat(132x192): col 0:64 = W0-W2, 64:128 = W1, 128:192 = 2*W2 ; rows>=130 zero.
__global__ void k_build_wcat(const float* __restrict__ gw, float* __restrict__ wc) {
  int idx = blockIdx.x * 256 + threadIdx.x;
  if (idx >= FP * MO) return;
  int f = idx / MO, c = idx % MO;
  float v = 0.f;
  if (f < 130) {
    int d = c & 63;
    if (c < 64)       v = gw[(f*3 + 0)*64 + d] - gw[(f*3 + 2)*64 + d];
    else if (c < 128) v = gw[(f*3 + 1)*64 + d];
    else              v = 2.f * gw[(f*3 + 2)*64 + d];
  }
  wc[idx] = v;
}

// Xc(BN x 132): f<2 inputs, 2..65 hx_k[:,5], 66..129 hx_k[:,4], pad 0
__global__ void k_build_xc(const float* __restrict__ inp, const float* __restrict__ hx,
                           float* __restrict__ xc) {
  int r = blockIdx.x;                 // 0..BN-1
  int bb = r >> 12, n = r & (Nn - 1);
  for (int f = threadIdx.x; f < FP; f += 64) {
    float v;
    if (f < 2)        v = inp[bb * (Nn * 2) + n * 2 + f];
    else if (f < 66)  v = hx[((size_t)(bb*6 + 5) * Nn + n) * Dd + (f - 2)];
    else if (f < 130) v = hx[((size_t)(bb*6 + 4) * Nn + n) * Dd + (f - 66)];
    else              v = 0.f;
    xc[(size_t)r * FP + f] = v;
  }
}

// ---------------- COO -> CSR ----------------

__global__ void k_zero_i32(int* __restrict__ a, int n) {
  int i = blockIdx.x * 256 + threadIdx.x;
  if (i < n) a[i] = 0;
}

__global__ void k_count(const int* __restrict__ rows, int* __restrict__ cnt) {
  int e = blockIdx.x * 256 + threadIdx.x;   // E = 512*256 exactly
  atomicAdd(&cnt[rows[e]], 1);
}

// one block, 1024 threads, exclusive scan of 4096 counts -> row_ptr; reset cursor
__global__ void k_scan(int* __restrict__ counts_and_cursor, int* __restrict__ row_ptr) {
  __shared__ int part[1024];
  int t = threadIdx.x;
  int c0 = counts_and_cursor[4*t + 0];
  int c1 = counts_and_cursor[4*t + 1];
  int c2 = counts_and_cursor[4*t + 2];
  int c3 = counts_and_cursor[4*t + 3];
  int s = c0 + c1 + c2 + c3;
  part[t] = s;
  __syncthreads();
  for (int off = 1; off < 1024; off <<= 1) {
    int v = (t >= off) ? part[t - off] : 0;
    __syncthreads();
    part[t] += v;
    __syncthreads();
  }
  int excl = part[t] - s;
  row_ptr[4*t + 0] = excl;
  row_ptr[4*t + 1] = excl + c0;
  row_ptr[4*t + 2] = excl + c0 + c1;
  row_ptr[4*t + 3] = excl + c0 + c1 + c2;
  counts_and_cursor[4*t + 0] = excl;
  counts_and_cursor[4*t + 1] = excl + c0;
  counts_and_cursor[4*t + 2] = excl + c0 + c1;
  counts_and_cursor[4*t + 3] = excl + c0 + c1 + c2;
  if (t == 1023) row_ptr[Nn] = part[1023];
}

__global__ void k_scatter(const int* __restrict__ rows, const int* __restrict__ cols,
                          const float* __restrict__ vals, int* __restrict__ cursor,
                          int* __restrict__ ccol, float* __restrict__ cval) {
  int e = blockIdx.x * 256 + threadIdx.x;
  int r = rows[e];
  int p = atomicAdd(&cursor[r], 1);
  ccol[p] = cols[e];
  cval[p] = vals[e];
}

// ---------------- GEMM1: [A0|A1|A2'] = Xc @ Wcat (f32 WMMA 16x16x4) ----------------
// block = 128 thr (4 waves), 16-row M tile; each wave owns 3 adjacent 16-col N tiles.
__global__ void k_gemm1(const float* __restrict__ xc, const float* __restrict__ wc,
                        float* __restrict__ abc) {
  __shared__ float As[16 * FP];        // bank stride 132 % 64 = 4 -> conflict-free cols
  const int t = threadIdx.x;
  const int m0 = blockIdx.x * 16;
  for (int idx = t; idx < 16 * FP; idx += 128)
    As[idx] = xc[(size_t)m0 * FP + idx];          // contiguous copy of the 16xFP tile
  __syncthreads();
  const int wave = t >> 5, lane = t & 31;
  const int half = lane >> 4, lm = lane & 15;
  const int koff = half * 2;                      // ISA f32 A/B layout: hi half-wave = K+2
  const int nb = wave * 48;
  v8f acc0 = {0.f,0.f,0.f,0.f,0.f,0.f,0.f,0.f};
  v8f acc1 = acc0, acc2 = acc0;
  for (int kk = 0; kk < FP; kk += 4) {
    v2f a = *(const v2f*)&As[lm * FP + kk + koff];            // A[M=lm, K=kk+koff..+1]
    const float* wr0 = wc + (size_t)(kk + koff    ) * MO + nb + lm;
    const float* wr1 = wc + (size_t)(kk + koff + 1) * MO + nb + lm;
    v2f b0 = { wr0[0],  wr1[0]  };
    v2f b1 = { wr0[16], wr1[16] };
    v2f b2 = { wr0[32], wr1[32] };
    acc0 = __builtin_amdgcn_wmma_f32_16x16x4_f32(false, a, false, b0, (short)0, acc0, false, false);
    acc1 = __builtin_amdgcn_wmma_f32_16x16x4_f32(false, a, false, b1, (short)0, acc1, false, false);
    acc2 = __builtin_amdgcn_wmma_f32_16x16x4_f32(false, a, false, b2, (short)0, acc2, false, false);
  }
  for (int i = 0; i < 8; ++i) {
    size_t row = (size_t)(m0 + i + 8 * half);     // C layout: VGPR i -> M = i + 8*half
    abc[row * MO + nb +      lm] = acc0[i];
    abc[row * MO + nb + 16 + lm] = acc1[i];
    abc[row * MO + nb + 32 + lm] = acc2[i];
  }
}

// ---------------- spmm passes (CSR rows, width 64, operands L2-resident) ----------------

// T = A1 + S * A2'
__global__ void k_spmm1(const float* __restrict__ abc, const int* __restrict__ row_ptr,
                        const int* __restrict__ ccol, const float* __restrict__ cval,
                        float* __restrict__ T) {
  const int n = blockIdx.x, bb = blockIdx.y, t = threadIdx.x;   // 64 threads
  __shared__ int sc[64];  __shared__ float sv[64];
  const int s = row_ptr[n], e = row_ptr[n + 1];
  float acc = 0.f;
  for (int base = s; base < e; base += 64) {
    int m = e - base; if (m > 64) m = 64;
    if (t < m) { sc[t] = ccol[base + t]; sv[t] = cval[base + t]; }
    __syncthreads();
    for (int j = 0; j < m; ++j)
      acc += sv[j] * abc[((size_t)(bb * Nn + sc[j])) * MO + 128 + t];   // A2' slice
    __syncthreads();
  }
  const size_t r = (size_t)(bb * Nn + n);
  T[r * 64 + t] = abc[r * MO + 64 + t] + acc;                           // + A1
}

// conv = leaky_relu(A0 + gconv_b + S * T)
__global__ void k_spmm2(const float* __restrict__ abc, const float* __restrict__ T,
                        const int* __restrict__ row_ptr, const int* __restrict__ ccol,
                        const float* __restrict__ cval, const float* __restrict__ gb,
                        float* __restrict__ ci) {
  const int n = blockIdx.x, bb = blockIdx.y, t = threadIdx.x;
  __shared__ int sc[64];  __shared__ float sv[64];
  const int s = row_ptr[n], e = row_ptr[n + 1];
  float acc = 0.f;
  for (int base = s; base < e; base += 64) {
    int m = e - base; if (m > 64) m = 64;
    if (t < m) { sc[t] = ccol[base + t]; sv[t] = cval[base + t]; }
    __syncthreads();
    for (int j = 0; j < m; ++j)
      acc += sv[j] * T[((size_t)(bb * Nn + sc[j])) * 64 + t];
    __syncthreads();
  }
  const size_t r = (size_t)(bb * Nn + n);
  float g = abc[r * MO + t] + gb[t] + acc;
  ci[r * 64 + t] = (g >= 0.f) ? g : 0.01f * g;
}

// ---------------- attention ----------------

// deterministic split reduction: partials[(bb*6+k)*32 + chunk]
__global__ void k_att_partial(const float* __restrict__ hx, const float* __restrict__ R,
                              const float* __restrict__ attw, float* __restrict__ parts) {
  const int bk = blockIdx.x;             // == bb*6 + k, indexes hx slab directly
  const int k = bk % 6;
  const int chunk = blockIdx.y;          // 32 chunks of 8192 over N*D
  const int t = threadIdx.x;             // 256
  const float* h = hx + (size_t)bk * (Nn * Dd);
  const float* r = R + (size_t)k * (Nn * Dd);
  const int base = chunk * 8192;
  float loc = 0.f;
  for (int i = t; i < 8192; i += 256)
    loc += (h[base + i] + r[base + i]) * attw[base + i];
  __shared__ float sdat[256];
  sdat[t] = loc;
  __syncthreads();
  for (int off = 128; off > 0; off >>= 1) {
    if (t < off) sdat[t] += sdat[t + off];
    __syncthreads();
  }
  if (t == 0) parts[bk * 32 + chunk] = sdat[0];
}

__global__ void k_softmax(const float* __restrict__ parts, const float* __restrict__ attb,
                          float* __restrict__ wgt) {
  int bb = threadIdx.x;
  if (bb >= Bb) return;
  float lg[6];
  for (int k = 0; k < 6; ++k) {
    float s = 0.f;
    for (int c = 0; c < 32; ++c) s += parts[(bb * 6 + k) * 32 + c];
    lg[k] = s + attb[0];
  }
  float mx = lg[0];
  for (int k = 1; k < 6; ++k) mx = fmaxf(mx, lg[k]);
  float den = 0.f;
  for (int k = 0; k < 6; ++k) { lg[k] = __expf(lg[k] - mx); den += lg[k]; }
  for (int k = 0; k < 6; ++k) wgt[bb * 6 + k] = lg[k] / den;
}

// ---------------- fused tail: out = conv@W + b + att ; hx shift ----------------
__global__ void k_final(const float* __restrict__ ci, const float* __restrict__ W,
                        const float* __restrict__ bmat, const float* __restrict__ hx,
                        const float* __restrict__ R, const float* __restrict__ wgt,
                        float* __restrict__ outp, float* __restrict__ hxnew) {
  __shared__ float As[16 * 68];          // pad 64->68 => bank stride 4, conflict-free
  const int t = threadIdx.x;
  const int m0 = blockIdx.x * 16;
  for (int idx = t; idx < 16 * 64; idx += 128) {
    int r = idx >> 6, c = idx & 63;
    As[r * 68 + c] = ci[(size_t)(m0 + r) * 64 + c];
  }
  __syncthreads();
  const int wave = t >> 5, lane = t & 31;
  const int half = lane >> 4, lm = lane & 15;
  const int koff = half * 2;
  const int n0 = wave * 16;
  v8f acc = {0.f,0.f,0.f,0.f,0.f,0.f,0.f,0.f};
  for (int kk = 0; kk < 64; kk += 4) {
    v2f a = *(const v2f*)&As[lm * 68 + kk + koff];
    v2f b = { W[(kk + koff) * 64 + n0 + lm], W[(kk + koff + 1) * 64 + n0 + lm] };
    acc = __builtin_amdgcn_wmma_f32_16x16x4_f32(false, a, false, b, (short)0, acc, false, false);
  }
  const int col = n0 + lm;
  const int bb = m0 >> 12;               // 16-row tiles never cross a batch boundary
  float wk[6];
  for (int k = 0; k < 6; ++k) wk[k] = wgt[bb * 6 + k];
  const size_t kstr = (size_t)Nn * 64;
  for (int i = 0; i < 8; ++i) {
    const int n = (m0 + i + 8 * half) & (Nn - 1);
    const size_t base = (size_t)bb * 6 * kstr + (size_t)n * 64 + col; // hx[bb,0,n,col]
    float att = 0.f;
    for (int k = 0; k < 6; ++k) {
      float h = hx[base + (size_t)k * kstr];
      att += wk[k] * (h + R[(size_t)k * kstr + (size_t)n * 64 + col]);
      if (k > 0) hxnew[base + (size_t)(k - 1) * kstr] = h;            // state shift
    }
    float ov = acc[i] + bmat[n * 64 + col] + att;
    outp[(size_t)bb * (Nn * 64) + (size_t)n * 64 + col] = ov;
    hxnew[base + 5 * kstr] = ov;
  }
}

// ---------------- launch ----------------

extern "C" void kernel_launch(void* const* d_in, const int* in_sizes, int n_in,
                              void* d_out, int out_size, void* d_ws, size_t ws_size,
                              hipStream_t stream) {
  (void)in_sizes; (void)n_in; (void)out_size; (void)ws_size;
  const float* inp  = (const float*)d_in[0];
  const float* hx   = (const float*)d_in[1];
  const int*   srow = (const int*)d_in[2];
  const int*   scol = (const int*)d_in[3];
  const float* sval = (const float*)d_in[4];
  const float* W    = (const float*)d_in[5];
  const float* bmat = (const float*)d_in[6];
  const float* R    = (const float*)d_in[7];
  const float* gw   = (const float*)d_in[8];
  const float* gb   = (const float*)d_in[9];
  const float* attw = (const float*)d_in[10];
  const float* attb = (const float*)d_in[11];

  float* outp  = (float*)d_out;
  float* hxnew = outp + (size_t)Bb * Nn * Dd;   // second tuple element

  char* p = (char*)d_ws;
  auto take = [&](size_t bytes) { char* q = p; p += (bytes + 255) & ~(size_t)255; return q; };
  float* abc  = (float*)take((size_t)BN * MO * 4);   // 100.7 MB
  float* xc   = (float*)take((size_t)BN * FP * 4);   // 69.2 MB (reused: T, conv)
  float* Tbuf = xc;                                  // alias, Xc dead after GEMM1
  float* ci   = xc + (size_t)BN * 64;                // alias, disjoint from Tbuf
  float* wc   = (float*)take((size_t)FP * MO * 4);
  int*   rp   = (int*)take((size_t)(Nn + 1) * 4);
  int*   cur  = (int*)take((size_t)Nn * 4);
  int*   ccol = (int*)take((size_t)Ee * 4);
  float* cval = (float*)take((size_t)Ee * 4);
  float* prt  = (float*)take((size_t)192 * 32 * 4);
  float* wgt  = (float*)take((size_t)192 * 4);

  k_build_wcat<<<(FP * MO + 255) / 256, 256, 0, stream>>>(gw, wc);
  k_build_xc<<<BN, 64, 0, stream>>>(inp, hx, xc);
  k_zero_i32<<<(Nn + 255) / 256, 256, 0, stream>>>(cur, Nn);
  k_count<<<Ee / 256, 256, 0, stream>>>(srow, cur);
  k_scan<<<1, 1024, 0, stream>>>(cur, rp);
  k_scatter<<<Ee / 256, 256, 0, stream>>>(srow, scol, sval, cur, ccol, cval);
  k_gemm1<<<BN / 16, 128, 0, stream>>>(xc, wc, abc);
  k_spmm1<<<dim3(Nn, Bb), 64, 0, stream>>>(abc, rp, ccol, cval, Tbuf);
  k_spmm2<<<dim3(Nn, Bb), 64, 0, stream>>>(abc, Tbuf, rp, ccol, cval, gb, ci);
  k_att_partial<<<dim3(192, 32), 256, 0, stream>>>(hx, R, attw, prt);
  k_softmax<<<1, 32, 0, stream>>>(prt, attb, wgt);
  k_final<<<BN / 16, 128, 0, stream>>>(ci, W, bmat, hx, R, wgt, outp, hxnew);
}